// _UniformEdgesSelector_6588479832170
// MI455X (gfx1250) — compile-verified
//
#include <hip/hip_runtime.h>
#include <hip/hip_bf16.h>
#include <stdint.h>

// ---------------------------------------------------------------------------
// Uniform without-replacement edge sampling (top-16 of 64 random keys per row)
// MI455X / gfx1250: TDM (tensor_load_to_lds) stages row tiles into LDS,
// threefry2x32 PRNG in VALU, branchless top-16 insertion network in VGPRs,
// b128 non-temporal output stores (outputs are write-once, 128 MB total).
// ---------------------------------------------------------------------------

#define N_ROWS          500000
#define HALF_ROWS       250000
#define MAX_DEG         64
#define S_SAMP          16
#define ROWS_PER_BLOCK  64      // row-PAIRS per block (block stages 2*64 rows)

typedef uint32_t u32;
typedef __attribute__((ext_vector_type(4))) unsigned int tdm_u32x4;
typedef __attribute__((ext_vector_type(4))) int          tdm_i32x4;
typedef __attribute__((ext_vector_type(8))) int          tdm_i32x8;
typedef __attribute__((ext_vector_type(4))) float        f32x4;

__device__ __forceinline__ u32 rotl32(u32 x, u32 r) {
  return (x << r) | (x >> (32u - r));
}

// threefry2x32 with key (0,0) == jax.random.key(0) key data
__device__ __forceinline__ void threefry2x32_key0(u32 x0, u32 x1, u32 &o0, u32 &o1) {
  const u32 ks2 = 0x1BD11BDAu;   // ks0 ^ ks1 ^ 0x1BD11BDA with ks0=ks1=0
  u32 v0 = x0, v1 = x1;          // initial key injection adds 0
#define TFR(r) v0 += v1; v1 = rotl32(v1, (r)); v1 ^= v0;
  TFR(13u) TFR(15u) TFR(26u) TFR(6u)
  /* inject 1 */ v1 += ks2 + 1u;
  TFR(17u) TFR(29u) TFR(16u) TFR(24u)
  /* inject 2 */ v0 += ks2; v1 += 2u;
  TFR(13u) TFR(15u) TFR(26u) TFR(6u)
  /* inject 3 */ v1 += 3u;
  TFR(17u) TFR(29u) TFR(16u) TFR(24u)
  /* inject 4 */ v1 += ks2 + 4u;
  TFR(13u) TFR(15u) TFR(26u) TFR(6u)
  /* inject 5 */ v0 += ks2; v1 += 5u;
#undef TFR
  o0 = v0; o1 = v1;
}

// Issue one 2-D TDM tile load: 64 cols x ROWS_PER_BLOCK rows of 4-byte data,
// row stride = 64 elements. tensor_rows clamps valid rows (OOB rows -> zero).
__device__ __forceinline__ void tdm_load_tile_2d(u32 lds_byte, const void *gptr,
                                                 u32 tensor_rows) {
  uint64_t ga = (uint64_t)(uintptr_t)gptr;
  tdm_u32x4 g0;
  g0[0] = 1u;                                            // count=1 (valid, user)
  g0[1] = lds_byte;                                      // lds_addr
  g0[2] = (u32)ga;                                       // global_addr[31:0]
  g0[3] = (u32)((ga >> 32) & 0x01FFFFFFu) | (2u << 30);  // addr[56:32] | type=2
  tdm_i32x8 g1;
  g1[0] = (int)(2u << 16);                               // data_size=2 -> 4 bytes
  g1[1] = (int)((u32)MAX_DEG << 16);                     // tensor_dim0 = 64
  g1[2] = (int)(tensor_rows << 16);                      // tensor_dim1[15:0]
  g1[3] = (int)((u32)MAX_DEG << 16);                     // tile_dim0 = 64
  g1[4] = (int)ROWS_PER_BLOCK;                           // tile_dim1 = 64 rows
  g1[5] = (int)MAX_DEG;                                  // dim0_stride = 64 elems
  g1[6] = 0;
  g1[7] = 0;
  tdm_i32x4 gz = {0, 0, 0, 0};
#if defined(__clang_major__) && (__clang_major__ >= 23)
  tdm_i32x8 gz8 = {0, 0, 0, 0, 0, 0, 0, 0};
  __builtin_amdgcn_tensor_load_to_lds(g0, g1, gz, gz, gz8, 0);
#else
  __builtin_amdgcn_tensor_load_to_lds(g0, g1, gz, gz, 0);
#endif
}

__global__ __launch_bounds__(ROWS_PER_BLOCK)
void uniform_edge_sampler(const float *__restrict__ edge_weights,
                          const int   *__restrict__ degrees,
                          const int   *__restrict__ edge_targets,
                          const int   *__restrict__ source_ids,
                          float       *__restrict__ out) {
  // 64 KB LDS: 2 row-groups x 64 rows x 64 cols x 4B for targets and weights
  __shared__ int   s_tgt[2][ROWS_PER_BLOCK * MAX_DEG];
  __shared__ float s_w  [2][ROWS_PER_BLOCK * MAX_DEG];

  const int t        = threadIdx.x;
  const int rowStart = blockIdx.x * ROWS_PER_BLOCK;

  // Wave 0 drives the TDM: 4 tile loads, then drain TENSORcnt.
  if (t < 32) {
    u32 rowsValid = (u32)(HALF_ROWS - rowStart);
    if (rowsValid > ROWS_PER_BLOCK) rowsValid = ROWS_PER_BLOCK;
    // low 32 bits of a flat LDS pointer == LDS byte offset (aperture layout)
    tdm_load_tile_2d((u32)(uintptr_t)&s_tgt[0][0],
                     edge_targets + (size_t)rowStart * MAX_DEG, rowsValid);
    tdm_load_tile_2d((u32)(uintptr_t)&s_tgt[1][0],
                     edge_targets + ((size_t)rowStart + HALF_ROWS) * MAX_DEG, rowsValid);
    tdm_load_tile_2d((u32)(uintptr_t)&s_w[0][0],
                     edge_weights + (size_t)rowStart * MAX_DEG, rowsValid);
    tdm_load_tile_2d((u32)(uintptr_t)&s_w[1][0],
                     edge_weights + ((size_t)rowStart + HALF_ROWS) * MAX_DEG, rowsValid);
    __builtin_amdgcn_s_wait_tensorcnt(0);
  }
  __syncthreads();

  // Each lane owns the row pair (pair, pair + 250000): one threefry block
  // yields one random key for each of the two rows (jax half-split pairing).
  const int  pair  = rowStart + t;
  const bool valid = pair < HALF_ROWS;
  const int  rA    = valid ? pair : 0;
  const int  rB    = rA + HALF_ROWS;

  const int degA = __builtin_nontemporal_load(degrees + rA);
  const int degB = __builtin_nontemporal_load(degrees + rB);

  // Running 16 smallest packed keys per row: key = (mantissa<<6) | column.
  u32 mA[S_SAMP], mB[S_SAMP];
#pragma unroll
  for (int j = 0; j < S_SAMP; ++j) { mA[j] = 0xFFFFFFFFu; mB[j] = 0xFFFFFFFFu; }

  const u32 baseF = (u32)(pair * MAX_DEG);
#pragma unroll 4
  for (int c = 0; c < MAX_DEG; ++c) {
    u32 rx, ry;
    threefry2x32_key0(baseF + (u32)c,
                      baseF + (u32)c + (u32)(HALF_ROWS * MAX_DEG), rx, ry);
    u32 kA = (c < degA) ? (((rx >> 9) << 6) | (u32)c) : (0xFF000000u | (u32)c);
    u32 kB = (c < degB) ? (((ry >> 9) << 6) | (u32)c) : (0xFF000000u | (u32)c);
    // Branchless sorted insertion (ascending); pushes out current max.
#pragma unroll
    for (int j = 0; j < S_SAMP; ++j) {
      u32 lo = kA < mA[j] ? kA : mA[j];
      u32 hi = kA < mA[j] ? mA[j] : kA;
      mA[j] = lo; kA = hi;
      u32 lo2 = kB < mB[j] ? kB : mB[j];
      u32 hi2 = kB < mB[j] ? mB[j] : kB;
      mB[j] = lo2; kB = hi2;
    }
  }

  if (valid) {
    const int nA = degA < S_SAMP ? degA : S_SAMP;
    const int nB = degB < S_SAMP ? degB : S_SAMP;
    const float srcA = (float)__builtin_nontemporal_load(source_ids + rA);
    const float srcB = (float)__builtin_nontemporal_load(source_ids + rB);
    const size_t NS = (size_t)N_ROWS * S_SAMP;   // per-output stride in d_out
    const size_t oA = (size_t)rA * S_SAMP;       // 64B-aligned row base
    const size_t oB = (size_t)rB * S_SAMP;
    const int ldsRow = t * MAX_DEG;

    // Assemble 4-wide chunks and emit b128 non-temporal stores
    // (outputs are streamed once; keep them out of L2's working set).
#pragma unroll
    for (int q = 0; q < S_SAMP / 4; ++q) {
      f32x4 vsA, vtA, vwA, vmA4;
      f32x4 vsB, vtB, vwB, vmB4;
#pragma unroll
      for (int e = 0; e < 4; ++e) {
        const int j  = q * 4 + e;
        const int cA = (int)(mA[j] & 63u);
        const int cB = (int)(mB[j] & 63u);
        const bool vA = j < nA;
        const bool vB = j < nB;
        vsA[e]  = vA ? srcA : 0.0f;
        vtA[e]  = vA ? (float)s_tgt[0][ldsRow + cA] : 0.0f;
        vwA[e]  = vA ? s_w[0][ldsRow + cA] : 0.0f;
        vmA4[e] = vA ? 1.0f : 0.0f;
        vsB[e]  = vB ? srcB : 0.0f;
        vtB[e]  = vB ? (float)s_tgt[1][ldsRow + cB] : 0.0f;
        vwB[e]  = vB ? s_w[1][ldsRow + cB] : 0.0f;
        vmB4[e] = vB ? 1.0f : 0.0f;
      }
      __builtin_nontemporal_store(vsA,  (f32x4*)(out + oA) + q);            // src
      __builtin_nontemporal_store(vtA,  (f32x4*)(out + NS + oA) + q);       // tgt
      __builtin_nontemporal_store(vwA,  (f32x4*)(out + 2 * NS + oA) + q);   // w
      __builtin_nontemporal_store(vmA4, (f32x4*)(out + 3 * NS + oA) + q);   // mask
      __builtin_nontemporal_store(vsB,  (f32x4*)(out + oB) + q);
      __builtin_nontemporal_store(vtB,  (f32x4*)(out + NS + oB) + q);
      __builtin_nontemporal_store(vwB,  (f32x4*)(out + 2 * NS + oB) + q);
      __builtin_nontemporal_store(vmB4, (f32x4*)(out + 3 * NS + oB) + q);
    }
  }
}

extern "C" void kernel_launch(void* const* d_in, const int* in_sizes, int n_in,
                              void* d_out, int out_size, void* d_ws, size_t ws_size,
                              hipStream_t stream) {
  (void)in_sizes; (void)n_in; (void)out_size; (void)d_ws; (void)ws_size;
  const float* edge_weights = (const float*)d_in[0];
  const int*   degrees      = (const int*)d_in[1];
  const int*   edge_targets = (const int*)d_in[2];
  const int*   source_ids   = (const int*)d_in[3];
  float*       out          = (float*)d_out;

  const int pairs  = HALF_ROWS;
  const int blocks = (pairs + ROWS_PER_BLOCK - 1) / ROWS_PER_BLOCK;  // 3907
  uniform_edge_sampler<<<dim3(blocks), dim3(ROWS_PER_BLOCK), 0, stream>>>(
      edge_weights, degrees, edge_targets, source_ids, out);
}